// WideAndDeepModel_83545703842150
// MI455X (gfx1250) — compile-verified
//
#include <hip/hip_runtime.h>
#include <hip/hip_bf16.h>

typedef float v2f __attribute__((ext_vector_type(2)));
typedef float v8f __attribute__((ext_vector_type(8)));

#define B_TOTAL   16384
#define DEEP_IN   164
#define N1        128
#define N2        64
#define N3        32
#define DIM2_W    10000

// Packed-weight workspace layout (float offsets into d_ws)
#define W1P_OFF   0          // 41*128*4 = 20992
#define W2P_OFF   20992      // 32*64*4  =  8192
#define W3P_OFF   29184      // 16*32*4  =  2048
#define B1P_OFF   31232      // 128
#define B2P_OFF   31360      // 64
#define B3P_OFF   31424      // 32
// total 31456 floats = 125824 bytes

#define L1_QUADS  (41 * 128) // 5248
#define L2_QUADS  (32 * 64)  // 2048
#define L3_QUADS  (16 * 32)  // 512
#define QUADS_END (L1_QUADS + L2_QUADS + L3_QUADS) // 7808
#define REPACK_WORK (QUADS_END + N1 + N2 + N3)     // 8032

// ---------------------------------------------------------------------------
// Pre-pass: repack W{1,2,3} into K-quad-interleaved layout with the BN affine
// folded in:  Wp[((k0/4)*N + n)*4 + j] = W[k0+j][n] * gamma[n]/sqrt(1+eps)
//             bp[n] = b[n]*gamma[n]/sqrt(1+eps) + beta[n]
// so the main kernel's B operand is one aligned b64 load per WMMA.
// ---------------------------------------------------------------------------
__global__ __launch_bounds__(256)
void WideAndDeepModel_repack_kernel(
    const float* __restrict__ W1, const float* __restrict__ b1,
    const float* __restrict__ W2, const float* __restrict__ b2,
    const float* __restrict__ W3, const float* __restrict__ b3,
    const float* __restrict__ g1, const float* __restrict__ be1,
    const float* __restrict__ g2, const float* __restrict__ be2,
    const float* __restrict__ g3, const float* __restrict__ be3,
    float* __restrict__ ws)
{
    const int tid = blockIdx.x * blockDim.x + threadIdx.x;
    if (tid >= REPACK_WORK) return;
    const float bnrs = rsqrtf(1.0f + 1e-5f);

    if (tid < L1_QUADS) {
        const int q = tid, kq = q >> 7, n = q & (N1 - 1);
        const float s = g1[n] * bnrs;
#pragma unroll
        for (int j = 0; j < 4; ++j)
            ws[W1P_OFF + (size_t)q * 4 + j] = W1[(size_t)(kq * 4 + j) * N1 + n] * s;
    } else if (tid < L1_QUADS + L2_QUADS) {
        const int q = tid - L1_QUADS, kq = q >> 6, n = q & (N2 - 1);
        const float s = g2[n] * bnrs;
#pragma unroll
        for (int j = 0; j < 4; ++j)
            ws[W2P_OFF + (size_t)q * 4 + j] = W2[(size_t)(kq * 4 + j) * N2 + n] * s;
    } else if (tid < QUADS_END) {
        const int q = tid - L1_QUADS - L2_QUADS, kq = q >> 5, n = q & (N3 - 1);
        const float s = g3[n] * bnrs;
#pragma unroll
        for (int j = 0; j < 4; ++j)
            ws[W3P_OFF + (size_t)q * 4 + j] = W3[(size_t)(kq * 4 + j) * N3 + n] * s;
    } else if (tid < QUADS_END + N1) {
        const int n = tid - QUADS_END;
        ws[B1P_OFF + n] = b1[n] * (g1[n] * bnrs) + be1[n];
    } else if (tid < QUADS_END + N1 + N2) {
        const int n = tid - QUADS_END - N1;
        ws[B2P_OFF + n] = b2[n] * (g2[n] * bnrs) + be2[n];
    } else {
        const int n = tid - QUADS_END - N1 - N2;
        ws[B3P_OFF + n] = b3[n] * (g3[n] * bnrs) + be3[n];
    }
}

// ---------------------------------------------------------------------------
// One FC layer (16 batch rows per wave) on the fp32 WMMA pipe:
//   out = relu(in @ Wp + bp)     (BN already folded into Wp/bp)
// in/out : LDS tiles; Wp packed [K/4][N][4]; one ds b64 + one global b64 per WMMA.
// ---------------------------------------------------------------------------
__device__ __forceinline__ void fc_wmma(
    const float* inbuf, int in_stride,
    float* outbuf, int out_stride,
    const float* __restrict__ Wp, const float* __restrict__ biasp,
    int K, int N, int lane)
{
    const int col = lane & 15;   // N-column of tile / M-row for A
    const int hi  = lane >> 4;   // 0: K pair (k0,k0+1), 1: (k0+2,k0+3)

    for (int n0 = 0; n0 < N; n0 += 16) {
        const int n = n0 + col;
        v8f acc = {0.f, 0.f, 0.f, 0.f, 0.f, 0.f, 0.f, 0.f};
        for (int k0 = 0; k0 < K; k0 += 4) {
            // A 16x4 fp32 layout (one b64 from LDS)
            const float* ap = inbuf + col * in_stride + k0 + 2 * hi;
            v2f a = *(const v2f*)ap;
            // B 4x16 fp32 layout (one b64 from packed weights)
            const float* bp = Wp + ((size_t)(k0 >> 2) * N + n) * 4 + 2 * hi;
            v2f b = *(const v2f*)bp;
            acc = __builtin_amdgcn_wmma_f32_16x16x4_f32(
                false, a, false, b, (short)0, acc, false, false);
        }
        const float bb = biasp[n];
#pragma unroll
        for (int j = 0; j < 8; ++j) {
            // C/D layout: VGPR j -> row j (lanes 0-15) / row j+8 (lanes 16-31)
            float v = acc[j] + bb;
            v = v > 0.f ? v : 0.f;
            outbuf[(j + 8 * hi) * out_stride + n] = v;
        }
    }
}

__global__ __launch_bounds__(64)
void WideAndDeepModel_83545703842150_kernel(
    const float* __restrict__ user, const float* __restrict__ price,
    const float* __restrict__ temporal,
    const float* __restrict__ item_emb, const float* __restrict__ cat_emb,
    const float* __restrict__ wide1, const float* __restrict__ wide2,
    const float* __restrict__ crossW,
    const int* __restrict__ item_ids, const int* __restrict__ cat_ids,
    const int* __restrict__ wf1, const int* __restrict__ wf2,
    const float* __restrict__ W4, const float* __restrict__ b4,
    const float* __restrict__ ws,
    float* __restrict__ out)
{
    // Two independent waves per block, each owning 16 batch rows.
    __shared__ float bufA[2][16 * DEEP_IN];  // x  -> h2
    __shared__ float bufB[2][16 * N1];       // h1 -> h3

    const int lane = threadIdx.x & 31;
    const int wave = threadIdx.x >> 5;
    float* A  = bufA[wave];
    float* Bb = bufB[wave];
    const int row0 = (blockIdx.x * 2 + wave) * 16;

    // ---- stage concat([user, item_emb[id], cat_emb[id], price, temporal]) ----
    for (int r = 0; r < 16; ++r) {
        const int grow = row0 + r;
        const int iid  = item_ids[grow];
        const int cid  = cat_ids[grow];
        float* dst = A + r * DEEP_IN;
        for (int c = lane; c < DEEP_IN; c += 32) {
            float v;
            if (c < 64)        v = user[(size_t)grow * 64 + c];
            else if (c < 128)  v = item_emb[(size_t)iid * 64 + (c - 64)];
            else if (c < 160)  v = cat_emb[(size_t)cid * 32 + (c - 128)];
            else if (c < 162)  v = price[(size_t)grow * 2 + (c - 160)];
            else               v = temporal[(size_t)grow * 2 + (c - 162)];
            dst[c] = v;
        }
    }
    __syncthreads();

    // ---- deep MLP on the fp32 WMMA pipe (BN folded into packed weights) ----
    fc_wmma(A, DEEP_IN, Bb, N1, ws + W1P_OFF, ws + B1P_OFF, DEEP_IN, N1, lane);
    __syncthreads();
    fc_wmma(Bb, N1, A, N2, ws + W2P_OFF, ws + B2P_OFF, N1, N2, lane);
    __syncthreads();
    fc_wmma(A, N2, Bb, N3, ws + W3P_OFF, ws + B3P_OFF, N2, N3, lane);
    __syncthreads();

    // ---- final 32->1 FC + wide component (scalar, lanes 0-15) ----
    if (lane < 16) {
        const int grow = row0 + lane;
        const float* h = Bb + lane * N3;
        float acc = b4[0];
#pragma unroll
        for (int k = 0; k < N3; ++k) acc += h[k] * W4[k];
        const int f1 = wf1[grow];
        const int f2 = wf2[grow];
        float wide = wide1[f1] + wide2[f2]
                   + crossW[(long long)f1 * DIM2_W + f2];
        out[grow] = acc + wide;
    }
}

extern "C" void kernel_launch(void* const* d_in, const int* in_sizes, int n_in,
                              void* d_out, int out_size, void* d_ws, size_t ws_size,
                              hipStream_t stream) {
    (void)in_sizes; (void)n_in; (void)out_size; (void)ws_size;
    float* ws = (float*)d_ws;

    // Pass 1: fold BN into weights/bias and repack for b64 B-operand loads.
    dim3 rgrid((REPACK_WORK + 255) / 256), rblock(256);
    WideAndDeepModel_repack_kernel<<<rgrid, rblock, 0, stream>>>(
        (const float*)d_in[12], (const float*)d_in[13],   // W1, b1
        (const float*)d_in[14], (const float*)d_in[15],   // W2, b2
        (const float*)d_in[16], (const float*)d_in[17],   // W3, b3
        (const float*)d_in[20], (const float*)d_in[21],   // g1, be1
        (const float*)d_in[22], (const float*)d_in[23],   // g2, be2
        (const float*)d_in[24], (const float*)d_in[25],   // g3, be3
        ws);

    // Pass 2: fused gather + WMMA MLP + wide component.
    dim3 grid(B_TOTAL / 32);   // 32 batch rows per block
    dim3 block(64);            // 2 waves (wave32)
    WideAndDeepModel_83545703842150_kernel<<<grid, block, 0, stream>>>(
        (const float*)d_in[0],  (const float*)d_in[1],  (const float*)d_in[2],
        (const float*)d_in[3],  (const float*)d_in[4],
        (const float*)d_in[5],  (const float*)d_in[6],  (const float*)d_in[7],
        (const int*)d_in[8],    (const int*)d_in[9],
        (const int*)d_in[10],   (const int*)d_in[11],
        (const float*)d_in[18], (const float*)d_in[19],  // W4, b4
        ws,
        (float*)d_out);
}